// CapsuleLayer_77575699300446
// MI455X (gfx1250) — compile-verified
//
#include <hip/hip_runtime.h>
#include <math.h>

// CapsNet dynamic routing, fully fused for gfx1250 (MI455X).
// u_hat (189MB) is never materialized: every round recomputes it with
// V_WMMA_F32_16X16X4_F32 in a TRANSPOSED tile [v x b].
//  - Round 1: routing weights softmax(bias) depend only on i, so they fold
//    into the A fragment -> a barrier-free chain of accumulating WMMAs.
//  - Rounds 2/3: agreement dot reduces over accumulator registers (in-lane)
//    plus one v_permlanex16 cross-half combine; softmax couples the 10
//    capsule-waves through LDS between two block barriers.

typedef float v2f __attribute__((ext_vector_type(2)));
typedef float v8f __attribute__((ext_vector_type(8)));

#define B_TOT   256
#define IN_CAPS 1152
#define IN_DIM  8
#define NUM_C   10
#define DIM_V   16
#define CV      160        // NUM_C * DIM_V
#define NBT     16         // 256 / 16 batch tiles
#define NCHUNK  18
#define CHUNK   64         // IN_CAPS / NCHUNK
#define WPAD    12

__device__ __forceinline__ float xhalf16(float v) {
    // v_permlanex16_b32, identity selectors: lane i of one 16-lane row reads
    // lane i of the other row (== shfl_xor mask 16, pure VALU, no LDS).
    int r = __builtin_amdgcn_permlanex16(__float_as_int(v), __float_as_int(v),
                                         0x76543210, (int)0xFEDCBA98, false, false);
    return __int_as_float(r);
}

__device__ __forceinline__ float pl16(float v, int s0, int s1) {
    int r = __builtin_amdgcn_permlane16(__float_as_int(v), __float_as_int(v),
                                        s0, s1, false, false);
    return __int_as_float(r);
}

// ---- Round-1 prep: rbias[i][c] = softmax_c(bias[i][:])  (batch-independent)
__global__ __launch_bounds__(256) void caps_bias_softmax(
    const float* __restrict__ bias, float* __restrict__ rbias)
{
    const int i = blockIdx.x * 256 + threadIdx.x;
    if (i >= IN_CAPS) return;
    float a[NUM_C];
#pragma unroll
    for (int c = 0; c < NUM_C; ++c) a[c] = bias[i * NUM_C + c];
    float mx = a[0];
#pragma unroll
    for (int c = 1; c < NUM_C; ++c) mx = fmaxf(mx, a[c]);
    float den = 0.f;
#pragma unroll
    for (int c = 0; c < NUM_C; ++c) { a[c] = __expf(a[c] - mx); den += a[c]; }
    const float inv = 1.f / den;
#pragma unroll
    for (int c = 0; c < NUM_C; ++c) rbias[i * NUM_C + c] = a[c] * inv;
}

// ---- Round 1: s1 = sum_i rbias[i][c] * u_hat  == pure accumulating GEMM.
// 320 threads = 10 waves; wave w owns capsule c = w. No barriers, no LDS.
__global__ __launch_bounds__(320) void caps_round1(
    const float* __restrict__ x,        // [256,1152,8]
    const float* __restrict__ Wt,       // [1152,10,8,16]
    const float* __restrict__ rbias,    // [1152,10]
    float* __restrict__ partial)        // [NBT*NCHUNK][16][160]
{
    const int bt   = blockIdx.x;
    const int ch   = blockIdx.y;
    const int w    = threadIdx.x >> 5;
    const int lane = threadIdx.x & 31;
    const int m    = lane & 15;          // col = batch row
    const int h    = lane >> 4;          // half: row v = r + 8h
    const int b0   = bt * 16;
    const int i0   = ch * CHUNK;

    v8f sacc = {0.f, 0.f, 0.f, 0.f, 0.f, 0.f, 0.f, 0.f};

    for (int ii = 0; ii < CHUNK; ++ii) {
        const int i = i0 + ii;
        const float rb = rbias[i * NUM_C + w];   // wave-uniform weight

        // A (16x4 f32): A[v][d] = rb * W[i][w][d][v]
        const float* wr = Wt + ((i * NUM_C + w) * IN_DIM) * DIM_V + m;
        v2f a1, a2;
        a1.x = rb * wr[(2 * h) * DIM_V];     a1.y = rb * wr[(2 * h + 1) * DIM_V];
        a2.x = rb * wr[(4 + 2 * h) * DIM_V]; a2.y = rb * wr[(5 + 2 * h) * DIM_V];

        // B (4x16 f32): B[d][b] = x[b0+b][i][d]
        const float* xr = x + (b0 + m) * (IN_CAPS * IN_DIM) + i * IN_DIM;
        v2f b1, b2;
        b1.x = xr[2 * h];     b1.y = xr[2 * h + 1];
        b2.x = xr[4 + 2 * h]; b2.y = xr[5 + 2 * h];

        // Accumulate directly in the WMMA C operand across all i
        sacc = __builtin_amdgcn_wmma_f32_16x16x4_f32(false, a1, false, b1,
                                                     (short)0, sacc, false, false);
        sacc = __builtin_amdgcn_wmma_f32_16x16x4_f32(false, a2, false, b2,
                                                     (short)0, sacc, false, false);
    }

    float* pp = partial + (size_t)(bt * NCHUNK + ch) * (16 * CV)
                        + m * CV + w * DIM_V + 8 * h;
    float4 s0 = {sacc[0], sacc[1], sacc[2], sacc[3]};
    float4 s1 = {sacc[4], sacc[5], sacc[6], sacc[7]};
    *(float4*)(pp)     = s0;
    *(float4*)(pp + 4) = s1;
}

// ---- Rounds 2/3: full routing iteration with agreement + softmax.
__global__ __launch_bounds__(320) void caps_route(
    const float* __restrict__ x,        // [256,1152,8]
    const float* __restrict__ Wt,       // [1152,10,8,16]
    const float* __restrict__ bias,     // [1152,10]
    const float* __restrict__ outPrev,  // [256,10,16]
    float* __restrict__ partial)        // [NBT*NCHUNK][16][160]
{
    __shared__ float ag[16][WPAD];            // routing logits [b-row][c]
    __shared__ float rw[16][WPAD];            // softmax weights [b-row][c]
    __shared__ float biasLds[CHUNK * NUM_C];  // bias slice for this i-chunk

    const int bt   = blockIdx.x;
    const int ch   = blockIdx.y;
    const int w    = threadIdx.x >> 5;
    const int lane = threadIdx.x & 31;
    const int m    = lane & 15;
    const int h    = lane >> 4;
    const int b0   = bt * 16;
    const int i0   = ch * CHUNK;

    for (int j = threadIdx.x; j < CHUNK * NUM_C; j += 320)
        biasLds[j] = bias[i0 * NUM_C + j];

    // Prev-output values matching the transposed C layout: reg r, half h ->
    // v = r+8h, batch = b0+m. Contiguous 8 floats -> two b128 loads.
    float ov[8];
    {
        const float* op = outPrev + (b0 + m) * CV + w * DIM_V + 8 * h;
        const float4 o0 = *(const float4*)(op);
        const float4 o1 = *(const float4*)(op + 4);
        ov[0] = o0.x; ov[1] = o0.y; ov[2] = o0.z; ov[3] = o0.w;
        ov[4] = o1.x; ov[5] = o1.y; ov[6] = o1.z; ov[7] = o1.w;
    }

    __syncthreads();   // biasLds visible

    v8f sacc = {0.f, 0.f, 0.f, 0.f, 0.f, 0.f, 0.f, 0.f};

    for (int ii = 0; ii < CHUNK; ++ii) {
        const int i = i0 + ii;

        const float* wr = Wt + ((i * NUM_C + w) * IN_DIM) * DIM_V + m;
        v2f a1, a2;
        a1.x = wr[(2 * h) * DIM_V];     a1.y = wr[(2 * h + 1) * DIM_V];
        a2.x = wr[(4 + 2 * h) * DIM_V]; a2.y = wr[(5 + 2 * h) * DIM_V];

        const float* xr = x + (b0 + m) * (IN_CAPS * IN_DIM) + i * IN_DIM;
        v2f b1, b2;
        b1.x = xr[2 * h];     b1.y = xr[2 * h + 1];
        b2.x = xr[4 + 2 * h]; b2.y = xr[5 + 2 * h];

        // u_hat^T tile [16v x 16b] for capsule w
        v8f uh = {0.f, 0.f, 0.f, 0.f, 0.f, 0.f, 0.f, 0.f};
        uh = __builtin_amdgcn_wmma_f32_16x16x4_f32(false, a1, false, b1,
                                                   (short)0, uh, false, false);
        uh = __builtin_amdgcn_wmma_f32_16x16x4_f32(false, a2, false, b2,
                                                   (short)0, uh, false, false);

        // logits: ag[b][w] = bias[i][w] + sum_v u_hat[b,w,v]*outPrev[b,w,v]
        {
            float t0 = fmaf(uh[0], ov[0], uh[1] * ov[1]);
            float t1 = fmaf(uh[2], ov[2], uh[3] * ov[3]);
            float t2 = fmaf(uh[4], ov[4], uh[5] * ov[5]);
            float t3 = fmaf(uh[6], ov[6], uh[7] * ov[7]);
            float p  = (t0 + t1) + (t2 + t3);
            p += xhalf16(p);
            if (lane < 16)
                ag[lane][w] = p + biasLds[ii * NUM_C + w];
        }
        __syncthreads();

        // softmax over c: lanes 0..15 handle one batch row each
        if (lane < 16) {
            float mx = ag[lane][0];
#pragma unroll
            for (int c = 1; c < NUM_C; ++c) mx = fmaxf(mx, ag[lane][c]);
            float den = 0.f;
#pragma unroll
            for (int c = 0; c < NUM_C; ++c)
                den += __expf(ag[lane][c] - mx);
            const float num = __expf(ag[lane][w] - mx);
            rw[lane][w] = num / den;
        }
        __syncthreads();

        const float rwt = rw[m][w];
#pragma unroll
        for (int r = 0; r < 8; ++r)
            sacc[r] = fmaf(rwt, uh[r], sacc[r]);
    }

    float* pp = partial + (size_t)(bt * NCHUNK + ch) * (16 * CV)
                        + m * CV + w * DIM_V + 8 * h;
    float4 s0 = {sacc[0], sacc[1], sacc[2], sacc[3]};
    float4 s1 = {sacc[4], sacc[5], sacc[6], sacc[7]};
    *(float4*)(pp)     = s0;
    *(float4*)(pp + 4) = s1;
}

// One block per batch element b; 160 threads = (c,v). Reduces the i-chunk
// partials, applies squash, optionally adds `addA` (for out1+out2 fusion).
__global__ __launch_bounds__(160) void caps_reduce(
    const float* __restrict__ partial,
    float* __restrict__ dst,
    const float* __restrict__ addA)
{
    const int b  = blockIdx.x;
    const int t  = threadIdx.x;        // c = t/16, v = t%16
    const int bt = b >> 4;
    const int br = b & 15;

    float val = 0.f;
#pragma unroll
    for (int ch = 0; ch < NCHUNK; ++ch)
        val += partial[(size_t)(bt * NCHUNK + ch) * (16 * CV) + br * CV + t];

    // |s|^2 over the 16 v-lanes of this capsule: permlane16 xor butterflies
    float sq = val * val;
    sq += pl16(sq, 0x67452301, (int)0xEFCDAB89);   // xor 1
    sq += pl16(sq, 0x54761032, (int)0xDCFE98BA);   // xor 2
    sq += pl16(sq, 0x32107654, (int)0xBA98FEDC);   // xor 4
    sq += pl16(sq, (int)0xFEDCBA98, 0x76543210);   // xor 8

    const float f = (sq / (1.f + sq)) / sqrtf(sq + 1e-7f);   // EPS = 1e-7
    float o = val * f;
    if (addA) o += addA[b * CV + t];
    dst[b * CV + t] = o;
}

extern "C" void kernel_launch(void* const* d_in, const int* in_sizes, int n_in,
                              void* d_out, int out_size, void* d_ws, size_t ws_size,
                              hipStream_t stream) {
    (void)in_sizes; (void)n_in; (void)out_size; (void)ws_size;
    const float* x    = (const float*)d_in[0];   // [256,1152,8]
    const float* Wt   = (const float*)d_in[1];   // [1152,10,8,16]
    const float* bias = (const float*)d_in[2];   // [1,1152,10,1] -> [1152,10]
    float* out = (float*)d_out;                  // [256,10,16]
    float* ws  = (float*)d_ws;

    float* out1    = ws;                          // 40960 floats
    float* o12     = ws + 40960;                  // 40960 floats (out1+out2)
    float* rbias   = ws + 81920;                  // 11520 floats
    float* partial = ws + 93440;                  // 737280 floats

    dim3 grid(NBT, NCHUNK), blk(320);

    // Round 1: weights = softmax(bias) (i-only) folded into the A fragment
    caps_bias_softmax<<<(IN_CAPS + 255) / 256, 256, 0, stream>>>(bias, rbias);
    caps_round1<<<grid, blk, 0, stream>>>(x, Wt, rbias, partial);
    caps_reduce<<<B_TOT, 160, 0, stream>>>(partial, out1, nullptr);

    // Round 2: logits = bias + <u_hat, out1>
    caps_route<<<grid, blk, 0, stream>>>(x, Wt, bias, out1, partial);
    caps_reduce<<<B_TOT, 160, 0, stream>>>(partial, o12, out1);  // o12 = out1 + out2

    // Round 3: logits = bias + <u_hat, out1+out2>
    caps_route<<<grid, blk, 0, stream>>>(x, Wt, bias, o12, partial);
    caps_reduce<<<B_TOT, 160, 0, stream>>>(partial, out, nullptr);
}